// EditLoss_55525337203377
// MI455X (gfx1250) — compile-verified
//
#include <hip/hip_runtime.h>

typedef __attribute__((ext_vector_type(2))) float v2f;
typedef __attribute__((ext_vector_type(8))) float v8f;

#define WAVE 32

#if __has_builtin(__builtin_amdgcn_wmma_f32_16x16x4_f32)
#define HAVE_WMMA_F32X4 1
#else
#define HAVE_WMMA_F32X4 0
#endif

// Pass 1: each wave reduces 64-element chunks.
// WMMA identity used: with B = ones, D[i][j] = sum_k A[i][k], so
// sum over all 256 C entries == 16 * sum(A tile), independent of VGPR layout.
// Two accumulators: cs accumulates x, cq accumulates x*x.
__global__ void pairmse_partials(const float* __restrict__ x, int n,
                                 float* __restrict__ partials, int totalWaves) {
    const int lane = threadIdx.x & (WAVE - 1);
    const int waveInBlock = threadIdx.x >> 5;
    const int wavesPerBlock = blockDim.x >> 5;
    const int waveId = blockIdx.x * wavesPerBlock + waveInBlock;

    v2f ones; ones.x = 1.0f; ones.y = 1.0f;
    v8f cs = {};  // running C matrix for sum(x)
    v8f cq = {};  // running C matrix for sum(x*x)

    const int nChunks = (n + 63) >> 6;   // 64 floats per wave per iteration
    for (int chunk = waveId; chunk < nChunks; chunk += totalWaves) {
        const int base = chunk << 6;
        const int i0 = base + lane;
        const int i1 = base + WAVE + lane;
        const float a0 = (i0 < n) ? x[i0] : 0.0f;
        const float a1 = (i1 < n) ? x[i1] : 0.0f;
#if HAVE_WMMA_F32X4
        v2f a;  a.x  = a0;      a.y  = a1;
        v2f aq; aq.x = a0 * a0; aq.y = a1 * a1;
        // D = A*ones + C : accumulates row sums of A into C (all 256 entries
        // together hold 16x the tile sum).
        cs = __builtin_amdgcn_wmma_f32_16x16x4_f32(
                false, a,  false, ones, (short)0, cs, false, false);
        cq = __builtin_amdgcn_wmma_f32_16x16x4_f32(
                false, aq, false, ones, (short)0, cq, false, false);
#else
        // VALU fallback: keep the same downstream semantics (16x scale).
        cs[0] += 16.0f * (a0 + a1);
        cq[0] += 16.0f * (a0 * a0 + a1 * a1);
#endif
    }

    float lsum = cs[0] + cs[1] + cs[2] + cs[3] + cs[4] + cs[5] + cs[6] + cs[7];
    float lsq  = cq[0] + cq[1] + cq[2] + cq[3] + cq[4] + cq[5] + cq[6] + cq[7];

    // wave32 tree reduction (deterministic order)
    #pragma unroll
    for (int off = 16; off >= 1; off >>= 1) {
        lsum += __shfl_xor(lsum, off, WAVE);
        lsq  += __shfl_xor(lsq,  off, WAVE);
    }

    if (lane == 0) {
        // undo the 16x from the all-ones WMMA (exact: power of two)
        partials[2 * waveId]     = lsum * 0.0625f;
        partials[2 * waveId + 1] = lsq  * 0.0625f;
    }
}

// Pass 2: single wave folds all per-wave partials and applies the closed form
//   mean_{i,j}(x_j - x_i)^2 = 2*S2/N - 2*(S1/N)^2
__global__ void pairmse_finalize(const float* __restrict__ partials,
                                 int totalWaves, float* __restrict__ out,
                                 float invN) {
    const int lane = threadIdx.x;
    float s = 0.0f, q = 0.0f;
    for (int i = lane; i < totalWaves; i += WAVE) {
        s += partials[2 * i];
        q += partials[2 * i + 1];
    }
    #pragma unroll
    for (int off = 16; off >= 1; off >>= 1) {
        s += __shfl_xor(s, off, WAVE);
        q += __shfl_xor(q, off, WAVE);
    }
    if (lane == 0) {
        const float mean = s * invN;
        out[0] = 2.0f * (q * invN) - 2.0f * mean * mean;
    }
}

extern "C" void kernel_launch(void* const* d_in, const int* in_sizes, int n_in,
                              void* d_out, int out_size, void* d_ws, size_t ws_size,
                              hipStream_t stream) {
    const float* x = (const float*)d_in[0];
    const int n = in_sizes[0];           // 16384
    float* out = (float*)d_out;          // 1 float
    float* partials = (float*)d_ws;      // 64 waves * 2 floats = 512 B

    const int threads = 256;             // 8 waves/block
    const int blocks = 8;                // 64 waves total; 256 elems/wave
    const int totalWaves = blocks * (threads / WAVE);

    pairmse_partials<<<blocks, threads, 0, stream>>>(x, n, partials, totalWaves);
    pairmse_finalize<<<1, WAVE, 0, stream>>>(partials, totalWaves, out,
                                             1.0f / (float)n);
}